// Pooling_84928683311564
// MI455X (gfx1250) — compile-verified
//
#include <hip/hip_runtime.h>
#include <hip/hip_bf16.h>

// ---------------------------------------------------------------------------
// GraphSAGE "pool" aggregator for MI455X (gfx1250):
//   h     = relu(feat @ W_pool + b_pool)          (WMMA f32 16x16x4)
//   neigh = segment_max(h[edge_src], edge_dst)    (u32 atomic max, values >= 0)
//   out   = max(neigh, 0) + feat                  (init-0 accumulator + add)
// ---------------------------------------------------------------------------

#define D_FEAT 128

typedef __attribute__((ext_vector_type(2))) float v2f;
typedef __attribute__((ext_vector_type(8))) float v8f;

// ---------------------------------------------------------------------------
// Kernel 1: h = relu(feat @ W + b), one wave32 per 16x16 output tile.
// A layout (16x4 f32):  lane&15 = M row; VGPR0/VGPR1 = K+0/K+1; lane>=16 -> K+2.
// B layout (4x16 f32):  lane&15 = N col; VGPR0/VGPR1 = K+0/K+1; lane>=16 -> K+2.
// C/D layout:           VGPR r -> M = r + 8*(lane>=16), N = lane&15.
// ---------------------------------------------------------------------------
__global__ void sage_gemm_relu_kernel(const float* __restrict__ feat,
                                      const float* __restrict__ W,
                                      const float* __restrict__ bias,
                                      float* __restrict__ h,
                                      int n_nodes, int n_row_tiles) {
    const int wave_id   = (blockIdx.x * blockDim.x + threadIdx.x) >> 5;
    const int lane      = threadIdx.x & 31;
    const int n_col_tiles = D_FEAT / 16;                  // 8
    const int row_tile  = wave_id / n_col_tiles;          // wave-uniform
    const int col_tile  = wave_id % n_col_tiles;
    if (row_tile >= n_row_tiles) return;                  // uniform branch: EXEC stays all-1

    const int m0    = row_tile * 16;
    const int n0    = col_tile * 16;
    const int lmod  = lane & 15;
    const int lhalf = lane >> 4;

    // Clamp row for loads (keeps every lane active for WMMA); guard stores below.
    int arow_idx = m0 + lmod;
    if (arow_idx >= n_nodes) arow_idx = n_nodes - 1;

    const float* __restrict__ arow = feat + (size_t)arow_idx * D_FEAT + lhalf * 2;
    const float* __restrict__ bcol = W + (size_t)(lhalf * 2) * D_FEAT + (n0 + lmod);

    v8f acc = {};
#pragma unroll
    for (int k = 0; k < D_FEAT; k += 4) {
        v2f a, b;
        a.x = arow[k + 0];                 // A[m][k + lhalf*2 + 0]
        a.y = arow[k + 1];                 // A[m][k + lhalf*2 + 1]
        b.x = bcol[(size_t)k * D_FEAT];            // W[k + lhalf*2 + 0][n]
        b.y = bcol[(size_t)k * D_FEAT + D_FEAT];   // W[k + lhalf*2 + 1][n]
        // (neg_a, A, neg_b, B, c_mod, C, reuse_a, reuse_b)
        acc = __builtin_amdgcn_wmma_f32_16x16x4_f32(false, a, false, b,
                                                    (short)0, acc, false, false);
    }

    const int   col = n0 + lmod;
    const float bv  = bias[col];
#pragma unroll
    for (int r = 0; r < 8; ++r) {
        const int row = m0 + r + lhalf * 8;
        if (row < n_nodes) {
            float v = acc[r] + bv;
            v = v > 0.0f ? v : 0.0f;       // relu
            h[(size_t)row * D_FEAT + col] = v;
        }
    }
}

// ---------------------------------------------------------------------------
// Kernel 2: zero the accumulator (bit pattern 0 == 0.0f). float4 stores.
// ---------------------------------------------------------------------------
__global__ void sage_zero_kernel(float4* __restrict__ out, int n4) {
    const int i = blockIdx.x * blockDim.x + threadIdx.x;
    if (i < n4) out[i] = make_float4(0.f, 0.f, 0.f, 0.f);
}

// ---------------------------------------------------------------------------
// Kernel 3: per-edge scatter-max. One wave per edge; each lane owns a float4
// chunk of the 128-wide feature row (512B coalesced per wave). Values are
// non-negative post-relu, so unsigned-integer atomicMax on the bit pattern is
// exact IEEE max.
// ---------------------------------------------------------------------------
__global__ void sage_edge_max_kernel(const int* __restrict__ edge_src,
                                     const int* __restrict__ edge_dst,
                                     const float* __restrict__ h,
                                     unsigned int* __restrict__ out,
                                     int n_edges) {
    const int e = (blockIdx.x * blockDim.x + threadIdx.x) >> 5;
    if (e >= n_edges) return;
    const int lane = threadIdx.x & 31;

    const int s = edge_src[e];             // wave-uniform -> scalarized
    const int d = edge_dst[e];

    const float4 hv = *reinterpret_cast<const float4*>(h + (size_t)s * D_FEAT + lane * 4);
    unsigned int* o = out + (size_t)d * D_FEAT + lane * 4;

    atomicMax(o + 0, __float_as_uint(hv.x));
    atomicMax(o + 1, __float_as_uint(hv.y));
    atomicMax(o + 2, __float_as_uint(hv.z));
    atomicMax(o + 3, __float_as_uint(hv.w));
}

// ---------------------------------------------------------------------------
// Kernel 4: out = neigh + feat  (float4 vectorized).
// ---------------------------------------------------------------------------
__global__ void sage_add_feat_kernel(const float4* __restrict__ feat,
                                     float4* __restrict__ out, int n4) {
    const int i = blockIdx.x * blockDim.x + threadIdx.x;
    if (i < n4) {
        float4 a = out[i];
        float4 f = feat[i];
        a.x += f.x; a.y += f.y; a.z += f.z; a.w += f.w;
        out[i] = a;
    }
}

extern "C" void kernel_launch(void* const* d_in, const int* in_sizes, int n_in,
                              void* d_out, int out_size, void* d_ws, size_t ws_size,
                              hipStream_t stream) {
    const float* feat     = (const float*)d_in[0];   // [N, 128]
    const float* W_pool   = (const float*)d_in[1];   // [128, 128]
    const float* b_pool   = (const float*)d_in[2];   // [128]
    const int*   edge_src = (const int*)d_in[3];     // [E]
    const int*   edge_dst = (const int*)d_in[4];     // [E]
    float*       out      = (float*)d_out;           // [N, 128]
    float*       h        = (float*)d_ws;            // scratch: [N, 128]

    const int n_nodes = in_sizes[0] / D_FEAT;        // 10000
    const int n_edges = in_sizes[3];                 // 640000

    // --- 1: h = relu(feat @ W + b), one wave per 16x16 tile --------------
    const int n_row_tiles = (n_nodes + 15) / 16;     // 625
    const int n_tiles     = n_row_tiles * (D_FEAT / 16);
    const int gemm_blocks = (n_tiles + 7) / 8;       // 8 waves / 256-thread block
    sage_gemm_relu_kernel<<<gemm_blocks, 256, 0, stream>>>(feat, W_pool, b_pool,
                                                           h, n_nodes, n_row_tiles);

    // --- 2: out = 0 ------------------------------------------------------
    const int n4 = (n_nodes * D_FEAT) / 4;
    sage_zero_kernel<<<(n4 + 255) / 256, 256, 0, stream>>>((float4*)out, n4);

    // --- 3: scatter-max over edges (one wave per edge) -------------------
    const int edge_blocks = (n_edges + 7) / 8;
    sage_edge_max_kernel<<<edge_blocks, 256, 0, stream>>>(edge_src, edge_dst, h,
                                                          (unsigned int*)out, n_edges);

    // --- 4: out += feat ---------------------------------------------------
    sage_add_feat_kernel<<<(n4 + 255) / 256, 256, 0, stream>>>((const float4*)feat,
                                                               (float4*)out, n4);
}